// GATVAEdecoder_41601053229532
// MI455X (gfx1250) — compile-verified
//
#include <hip/hip_runtime.h>
#include <math.h>
#include <stdint.h>

#define B_    256
#define N_    256
#define FIN   768
#define H_    4
#define FOUT  192
#define NEGV  (-1e9f)
#define SLOPE 0.2f

typedef __attribute__((ext_vector_type(2))) float v2f;
typedef __attribute__((ext_vector_type(8))) float v8f;

// ---------------------------------------------------------------------------
// CDNA5 async global->LDS copy (16B per lane, ASYNCcnt-tracked), with a
// synchronous fallback if the builtin is absent on this toolchain.
// Builtin signature (from hipcc diagnostic): (v4i AS1*, v4i AS3*, imm, imm).
// ---------------------------------------------------------------------------
#if __has_builtin(__builtin_amdgcn_global_load_async_to_lds_b128)
#define HAVE_ASYNC_LDS 1
typedef int v4i_t __attribute__((vector_size(16)));
typedef __attribute__((address_space(1))) v4i_t GV4;
typedef __attribute__((address_space(3))) v4i_t LV4;
__device__ __forceinline__ void async_copy16(const void* g, void* l) {
  __builtin_amdgcn_global_load_async_to_lds_b128(
      (GV4*)(uintptr_t)g, (LV4*)(uint32_t)(uintptr_t)l, 0, 0);
}
#else
#define HAVE_ASYNC_LDS 0
__device__ __forceinline__ void async_copy16(const void* g, void* l) {
  *(float4*)l = *(const float4*)g;   // sync fallback: global load + ds_store
}
#endif

__device__ __forceinline__ void wait_async0() {
#if HAVE_ASYNC_LDS
#if __has_builtin(__builtin_amdgcn_s_wait_asynccnt)
  __builtin_amdgcn_s_wait_asynccnt(0);
#else
  asm volatile("s_wait_asynccnt 0" ::: "memory");
#endif
#endif
}

__device__ __forceinline__ v8f wmma_f32(v2f a, v2f b, v8f c) {
  return __builtin_amdgcn_wmma_f32_16x16x4_f32(false, a, false, b,
                                               (short)0, c, false, false);
}

// ---------------------------------------------------------------------------
// Kernel A: h'[b,h,n,o] = sum_f x[b,n,f] * w[h,f,o]
// WG = 256 threads (8 waves) computes 128(M) x 64(O); K double-buffered in
// 32-chunks via async-to-LDS. Each wave register-blocks a 2x2 grid of 16x16
// tiles: 4 fragment loads feed 4 v_wmma_f32_16x16x4_f32 per k-step.
// ---------------------------------------------------------------------------
__global__ __launch_bounds__(256) void gat_proj(const float* __restrict__ x,
                                                const float* __restrict__ w,
                                                float* __restrict__ hp) {
  __shared__ float Xs[2][128 * 36];  // stride 36: 16B-aligned rows, no A-frag conflicts
  __shared__ float Ws[2][32 * 68];   // stride 68: 16B-aligned rows

  const int tile = blockIdx.x;              // 0..5 = 2 m-tiles x 3 o-tiles
  const int h    = blockIdx.y;
  const int b    = blockIdx.z;
  const int m0   = (tile & 1) * 128;
  const int o0   = (tile >> 1) * 64;

  const int tid  = threadIdx.x;
  const int lane = tid & 31, wave = tid >> 5;
  const int wm   = wave >> 1, wn = wave & 1;   // 4x2 wave grid of 32x32 blocks
  const int m16  = lane & 15, kh = lane >> 4;  // WMMA lane decomposition

  const float* xb = x + (size_t)b * N_ * FIN;
  const float* wh = w + (size_t)h * FIN * FOUT;

  auto stage = [&](int k0, int bf) {
#pragma unroll
    for (int j = 0; j < 4; ++j) {            // X chunk 128x32 = 1024 float4
      int idx = tid + j * 256;
      int r = idx >> 3, c = (idx & 7) * 4;
      async_copy16(&xb[(size_t)(m0 + r) * FIN + k0 + c], &Xs[bf][r * 36 + c]);
    }
#pragma unroll
    for (int j = 0; j < 2; ++j) {            // W chunk 32x64 = 512 float4
      int idx = tid + j * 256;
      int r = idx >> 4, c = (idx & 15) * 4;
      async_copy16(&wh[(size_t)(k0 + r) * FOUT + o0 + c], &Ws[bf][r * 68 + c]);
    }
  };

  v8f a00 = {}, a01 = {}, a10 = {}, a11 = {};

  stage(0, 0);
  wait_async0();
  __syncthreads();

  int bf = 0;
  for (int ch = 0; ch < FIN / 32; ++ch) {
    if (ch + 1 < FIN / 32) stage((ch + 1) * 32, bf ^ 1);  // overlap with WMMA
    const float* Xb = &Xs[bf][0];
    const float* Wb = &Ws[bf][0];
#pragma unroll
    for (int kk = 0; kk < 32; kk += 4) {
      int ka = kk + kh * 2;
      v2f fa0, fa1, fb0, fb1;
      fa0.x = Xb[(wm * 32 + m16) * 36 + ka];
      fa0.y = Xb[(wm * 32 + m16) * 36 + ka + 1];
      fa1.x = Xb[(wm * 32 + 16 + m16) * 36 + ka];
      fa1.y = Xb[(wm * 32 + 16 + m16) * 36 + ka + 1];
      fb0.x = Wb[ka * 68 + wn * 32 + m16];
      fb0.y = Wb[(ka + 1) * 68 + wn * 32 + m16];
      fb1.x = Wb[ka * 68 + wn * 32 + 16 + m16];
      fb1.y = Wb[(ka + 1) * 68 + wn * 32 + 16 + m16];
      a00 = wmma_f32(fa0, fb0, a00);
      a01 = wmma_f32(fa0, fb1, a01);
      a10 = wmma_f32(fa1, fb0, a10);
      a11 = wmma_f32(fa1, fb1, a11);
    }
    wait_async0();     // my prefetch for chunk ch+1 has landed
    __syncthreads();   // everyone done reading buf & done writing buf^1
    bf ^= 1;
  }

  float* hpo = hp + (((size_t)b * H_ + h) * N_) * FOUT;
#pragma unroll
  for (int r = 0; r < 8; ++r) {
    int mr = m0 + wm * 32 + r + 8 * kh;
    int oc = o0 + wn * 32 + m16;
    hpo[(size_t)mr * FOUT + oc]             = a00[r];
    hpo[(size_t)mr * FOUT + oc + 16]        = a01[r];
    hpo[(size_t)(mr + 16) * FOUT + oc]      = a10[r];
    hpo[(size_t)(mr + 16) * FOUT + oc + 16] = a11[r];
  }
}

// ---------------------------------------------------------------------------
// Kernel B: fused scores + masked softmax + aggregation + ELU, one WG/(b,h).
// h'[b,h] (256x192 f32) staged in LDS via async-to-LDS (320KB/WGP).
// Bands of 32 rows: softmax (4 rows/wave), then attn(32x256) x h'(256x192)
// with each wave doing 3 o-tiles that share one A-fragment per k-step.
// ---------------------------------------------------------------------------
__global__ __launch_bounds__(256) void gat_attn(const float* __restrict__ hp,
                                                const int*   __restrict__ adj,
                                                const float* __restrict__ Wm,
                                                const float* __restrict__ a_src,
                                                const float* __restrict__ a_dst,
                                                const float* __restrict__ bias,
                                                float* __restrict__ out) {
  extern __shared__ float smem[];
  const int HPS = 196;                        // padded h' row stride (16B-aligned)
  float* hS    = smem;                        // 256*196
  float* attnS = hS + N_ * HPS;               // 32*257 (odd stride: conflict-free A-frag)
  float* sSrc  = attnS + 32 * 257;            // 256
  float* sDst  = sSrc + N_;                   // 256
  float* aS    = sDst + N_;                   // 192
  float* adS   = aS + FOUT;                   // 192
  float* bS    = adS + FOUT;                  // 192

  const int h    = blockIdx.x;
  const int b    = blockIdx.y;
  const int tid  = threadIdx.x;
  const int lane = tid & 31, wave = tid >> 5;
  const int m16  = lane & 15, kh = lane >> 4;

  const float* hpb = hp + (((size_t)b * H_ + h) * N_) * FOUT;

  // 1) stage h' via async-to-LDS (48 x 16B per thread, coalesced)
  for (int i = tid; i < N_ * FOUT / 4; i += 256) {
    int e = i * 4;
    int r = e / FOUT, c = e - r * FOUT;
    async_copy16(&hpb[(size_t)r * FOUT + c], &hS[r * HPS + c]);
  }
  for (int i = tid; i < FOUT; i += 256) {
    aS[i]  = a_src[h * FOUT + i];
    adS[i] = a_dst[h * FOUT + i];
    bS[i]  = bias[i];
  }
  wait_async0();
  __syncthreads();

  // 2) s_src/s_dst: one row per thread out of LDS
  {
    float s0 = 0.f, s1 = 0.f;
    const float* row = &hS[tid * HPS];
    for (int o = 0; o < FOUT; ++o) {
      float v = row[o];
      s0 += v * aS[o];
      s1 += v * adS[o];
    }
    sSrc[tid] = s0;
    sDst[tid] = s1;
  }
  __syncthreads();

  const float* Wb = Wm + (size_t)b * N_ * N_;
  const int*   Ab = adj + (size_t)b * N_ * N_;

  const int mh  = wave & 1;     // which 16-row half of the band this wave owns
  const int ot0 = wave >> 1;    // o-tiles: ot0, ot0+4, ot0+8

  for (int band = 0; band < 8; ++band) {
    // a) masked logits + softmax: 4 rows per wave
#pragma unroll
    for (int rr = 0; rr < 4; ++rr) {
      int ml = wave * 4 + rr;
      int m  = band * 32 + ml;
      if (band + 1 < 8) {   // warm next band's W/adj rows (global_prefetch_b8)
        int mn = (band + 1) * 32 + ml;
        __builtin_prefetch(&Wb[(size_t)mn * N_ + lane * 8], 0, 0);
        __builtin_prefetch(&Ab[(size_t)mn * N_ + lane * 8], 0, 0);
      }
      float ss = sSrc[m];
      float lg[8];
      float mx = -3.4e38f;
#pragma unroll
      for (int i = 0; i < 8; ++i) {
        int j   = lane + i * 32;                 // coalesced over lanes
        float l = ss + sDst[j];
        l = (l >= 0.f) ? l : SLOPE * l;          // leaky_relu(0.2)
        l += Wb[(size_t)m * N_ + j];
        l = (Ab[(size_t)m * N_ + j] != 0) ? l : NEGV;
        lg[i] = l;
        mx = fmaxf(mx, l);
      }
      for (int off = 16; off > 0; off >>= 1) mx = fmaxf(mx, __shfl_xor(mx, off, 32));
      float sum = 0.f;
#pragma unroll
      for (int i = 0; i < 8; ++i) {
        lg[i] = expf(lg[i] - mx);
        sum += lg[i];
      }
      for (int off = 16; off > 0; off >>= 1) sum += __shfl_xor(sum, off, 32);
      float inv = 1.f / sum;
#pragma unroll
      for (int i = 0; i < 8; ++i) attnS[ml * 257 + lane + i * 32] = lg[i] * inv;
    }
    __syncthreads();

    // b) aggregation: out[32 x 192] = attn_band(32x256) * h'(256x192)
    //    wave computes 3 tiles (same 16-row half): 1 A-frag feeds 3 WMMAs
    v8f c0 = {}, c1 = {}, c2 = {};
    for (int k = 0; k < N_; k += 4) {
      int ka = k + kh * 2;
      v2f fa, fb0, fb1, fb2;
      fa.x  = attnS[(mh * 16 + m16) * 257 + ka];
      fa.y  = attnS[(mh * 16 + m16) * 257 + ka + 1];
      fb0.x = hS[ka * HPS + ot0 * 16 + m16];
      fb0.y = hS[(ka + 1) * HPS + ot0 * 16 + m16];
      fb1.x = hS[ka * HPS + (ot0 + 4) * 16 + m16];
      fb1.y = hS[(ka + 1) * HPS + (ot0 + 4) * 16 + m16];
      fb2.x = hS[ka * HPS + (ot0 + 8) * 16 + m16];
      fb2.y = hS[(ka + 1) * HPS + (ot0 + 8) * 16 + m16];
      c0 = wmma_f32(fa, fb0, c0);
      c1 = wmma_f32(fa, fb1, c1);
      c2 = wmma_f32(fa, fb2, c2);
    }
#pragma unroll
    for (int r = 0; r < 8; ++r) {
      int n = band * 32 + mh * 16 + r + 8 * kh;
      size_t base = ((size_t)b * N_ + n) * (H_ * FOUT) + (size_t)h * FOUT;
      float v0 = c0[r] + bS[ot0 * 16 + m16];
      float v1 = c1[r] + bS[(ot0 + 4) * 16 + m16];
      float v2 = c2[r] + bS[(ot0 + 8) * 16 + m16];
      v0 = (v0 > 0.f) ? v0 : expm1f(v0);        // ELU
      v1 = (v1 > 0.f) ? v1 : expm1f(v1);
      v2 = (v2 > 0.f) ? v2 : expm1f(v2);
      out[base + ot0 * 16 + m16]        = v0;
      out[base + (ot0 + 4) * 16 + m16]  = v1;
      out[base + (ot0 + 8) * 16 + m16]  = v2;
    }
    __syncthreads();
  }
}

// ---------------------------------------------------------------------------
extern "C" void kernel_launch(void* const* d_in, const int* in_sizes, int n_in,
                              void* d_out, int out_size, void* d_ws, size_t ws_size,
                              hipStream_t stream) {
  const float* x     = (const float*)d_in[0];
  /* d_in[1] = doc_len: unused by the reference */
  const int*   adj   = (const int*)  d_in[2];
  const float* Wm    = (const float*)d_in[3];
  const float* w     = (const float*)d_in[4];
  const float* a_src = (const float*)d_in[5];
  const float* a_dst = (const float*)d_in[6];
  const float* bias  = (const float*)d_in[7];
  float* out = (float*)d_out;
  float* hp  = (float*)d_ws;   // B*H*N*FOUT f32 = 201 MB scratch

  dim3 gA(6, H_, B_);
  gat_proj<<<gA, 256, 0, stream>>>(x, w, hp);

  size_t lds = (size_t)(N_ * 196 + 32 * 257 + 2 * N_ + 3 * FOUT) * sizeof(float);
  dim3 gC(H_, B_);
  gat_attn<<<gC, 256, lds, stream>>>(hp, adj, Wm, a_src, a_dst, bias, out);
}